// GlobalNHC_46600395162250
// MI455X (gfx1250) — compile-verified
//
#include <hip/hip_runtime.h>

typedef __attribute__((ext_vector_type(2))) float v2f;
typedef __attribute__((ext_vector_type(8))) float v8f;

#define TPB 256
#define CHUNK 4096  // elements of one batch handled per block

// ---------------------------------------------------------------------------
// Kernel A: per-batch kinetic-energy partial reduction.
// grid = (G, B). Each block reduces CHUNK elements of batch blockIdx.y to one
// float partial. Wave reduction uses v_wmma_f32_16x16x4_f32 for the
// lane<->lane+16 fold (A = ones, B = partials; D[m][n] = p_n + p_{n+16}),
// then xor-shuffles for the remaining 16-wide fold.
// ---------------------------------------------------------------------------
__global__ void nhc_reduce_kernel(const float* __restrict__ mom,
                                  const float* __restrict__ mas,
                                  float* __restrict__ partials,
                                  int ND, int N, int D, int G) {
  const int b = blockIdx.y;
  const float* momb = mom + (long)b * ND;
  const float* masb = mas + (long)b * N;
  const int start = blockIdx.x * CHUNK;

  float p = 0.0f;
  #pragma unroll
  for (int k = 0; k < CHUNK; k += TPB) {
    int e = start + k + threadIdx.x;
    if (e < ND) {
      float m = momb[e];
      float q = masb[e / D];
      p += m * m / q;
    }
  }

  // --- wave32 reduction, step 1 via WMMA (EXEC is full here: no divergence) ---
  v2f a; a.x = 1.0f; a.y = 1.0f;     // A(16x4) = all ones
  v2f bm; bm.x = p;  bm.y = 0.0f;    // B(4x16): one row-pair holds partials
  v8f c = {};
  v8f d = __builtin_amdgcn_wmma_f32_16x16x4_f32(false, a, false, bm,
                                                (short)0, c, false, false);
  float s = d[0];                    // lane i: p_{i%16} + p_{i%16+16}
  s += __shfl_xor(s, 1, 32);
  s += __shfl_xor(s, 2, 32);
  s += __shfl_xor(s, 4, 32);
  s += __shfl_xor(s, 8, 32);         // full 32-lane sum in every lane

  __shared__ float wsum[TPB / 32];
  const int lane = threadIdx.x & 31;
  const int wave = threadIdx.x >> 5;
  if (lane == 0) wsum[wave] = s;
  __syncthreads();
  if (threadIdx.x == 0) {
    float t = 0.0f;
    #pragma unroll
    for (int i = 0; i < TPB / 32; ++i) t += wsum[i];
    partials[(long)b * G + blockIdx.x] = t;
  }
}

// ---------------------------------------------------------------------------
// Kernel B: one wave; thread b integrates the 5-element NHC chain through all
// NRESPA*7 substeps, tracking KE analytically (KE *= s^2) and the cumulative
// atomic-momentum scale S[b]. Mirrors the reference's f32 arithmetic.
// ---------------------------------------------------------------------------
__global__ void nhc_chain_kernel(const float* __restrict__ partials, int G,
                                 const float* __restrict__ kbt,
                                 const float* __restrict__ dtm,
                                 const float* __restrict__ pos_nhc_in,
                                 const float* __restrict__ mom_nhc_in,
                                 const float* __restrict__ mas_nhc,
                                 const float* __restrict__ stp_p,
                                 int Bn, int C, float dof,
                                 float* __restrict__ outPos,
                                 float* __restrict__ outMomNhc,
                                 float* __restrict__ Sout) {
  const int b = threadIdx.x;
  if (b >= Bn || C > 8) return;

  double acc = 0.0;
  for (int g = 0; g < G; ++g) acc += (double)partials[(long)b * G + g];
  float KE = (float)acc;

  const float kb  = kbt[b];
  const float dt  = dtm[b];
  const float stp = stp_p[0];

  float pn[8], pc[8], mn[8];
  for (int j = 0; j < C; ++j) {
    pn[j] = pos_nhc_in[b * C + j];
    pc[j] = mom_nhc_in[b * C + j];
    mn[j] = mas_nhc[b * C + j];
  }

  const double W1 = 0.78451361047756;
  const double W2 = 0.235573213359357;
  const double W3 = -1.17767998417887;
  const double W4 = 1.0 - 2.0 * (W1 + W2 + W3);
  const double W[7] = {W1, W2, W3, W4, W3, W2, W1};

  float S = 1.0f;
  for (int r = 0; r < 2; ++r) {          // NRESPA
    for (int iw = 0; iw < 7; ++iw) {
      const float w   = (float)W[iw];
      const float dea = dt * (stp * w * 0.5f);   // dtm * (stp*w/NRESPA)
      float g[8], mc[8];

      // g from KE and substep-start chain momenta
      g[0] = KE - kb * dof;
      for (int j = 1; j < C; ++j) g[j] = pc[j - 1] * pc[j - 1] / mn[j - 1] - kb;
      for (int j = 0; j < C; ++j) mc[j] = pc[j];

      mc[C - 1] += g[C - 1] * (dea * 0.5f);
      for (int j = C - 2; j >= 0; --j) {
        float f = expf(-mc[j + 1] / mn[j + 1] * (dea * 0.25f));
        mc[j] = (mc[j] * f + g[j] * (dea * 0.5f)) * f;
      }

      for (int j = 0; j < C; ++j) pn[j] += mc[j] / mn[j] * dea;

      const float s = expf(-mc[0] / mn[0] * dea);
      S *= s;
      KE *= s * s;

      // second half: g2 identical except KE-derived g[0]
      g[0] = KE - kb * dof;
      for (int j = C - 2; j >= 0; --j) {
        float f = expf(-mc[j + 1] / mn[j + 1] * (dea * 0.25f));
        mc[j] = (mc[j] * f + g[j] * (dea * 0.5f)) * f;
      }
      mc[C - 1] += g[C - 1] * (dea * 0.5f);

      for (int j = 0; j < C; ++j) pc[j] = mc[j];
    }
  }

  for (int j = 0; j < C; ++j) {
    outPos[b * C + j]    = pn[j];
    outMomNhc[b * C + j] = pc[j];
  }
  Sout[b] = S;
}

// ---------------------------------------------------------------------------
// Kernel C: mom_out = mom_in * S[batch], float4-vectorized streaming pass.
// ---------------------------------------------------------------------------
__global__ void nhc_scale_vec4(const float4* __restrict__ momIn,
                               float4* __restrict__ momOut,
                               const float* __restrict__ S,
                               int vecPerBatch, long totalVec) {
  long i = (long)blockIdx.x * blockDim.x + threadIdx.x;
  if (i >= totalVec) return;
  const float s = S[(int)(i / vecPerBatch)];
  float4 v = momIn[i];
  v.x *= s; v.y *= s; v.z *= s; v.w *= s;
  momOut[i] = v;
}

__global__ void nhc_scale_scalar(const float* __restrict__ momIn,
                                 float* __restrict__ momOut,
                                 const float* __restrict__ S,
                                 int ND, long total) {
  long i = (long)blockIdx.x * blockDim.x + threadIdx.x;
  if (i >= total) return;
  momOut[i] = momIn[i] * S[(int)(i / ND)];
}

// ---------------------------------------------------------------------------
extern "C" void kernel_launch(void* const* d_in, const int* in_sizes, int n_in,
                              void* d_out, int out_size, void* d_ws, size_t ws_size,
                              hipStream_t stream) {
  // setup_inputs order: pos, mom, mas, kbt, dtm, pos_nhc, mom_nhc, mas_nhc, stp
  const float* mom      = (const float*)d_in[1];
  const float* mas      = (const float*)d_in[2];
  const float* kbt      = (const float*)d_in[3];
  const float* dtm      = (const float*)d_in[4];
  const float* pos_nhc  = (const float*)d_in[5];
  const float* mom_nhc  = (const float*)d_in[6];
  const float* mas_nhc  = (const float*)d_in[7];
  const float* stp      = (const float*)d_in[8];

  const int B  = in_sizes[3];           // kbt is [B]
  const int ND = in_sizes[1] / B;       // N*D per batch
  const int N  = in_sizes[2] / B;       // mas is [B,N]
  const int D  = ND / N;
  const int C  = in_sizes[5] / B;       // pos_nhc is [B,C]
  const int G  = (ND + CHUNK - 1) / CHUNK;

  float* partials = (float*)d_ws;               // B*G floats
  float* Sfac     = partials + (size_t)B * G;   // B floats

  float* outMom  = (float*)d_out;
  float* outPos  = outMom + (size_t)B * ND;
  float* outMomN = outPos + (size_t)B * C;

  // 1) per-batch KE partials
  dim3 gridA(G, B);
  nhc_reduce_kernel<<<gridA, TPB, 0, stream>>>(mom, mas, partials, ND, N, D, G);

  // 2) chain integration (one wave; thread per batch)
  int bt = ((B + 31) / 32) * 32;
  if (bt > 1024) bt = 1024;
  nhc_chain_kernel<<<1, bt, 0, stream>>>(partials, G, kbt, dtm, pos_nhc,
                                         mom_nhc, mas_nhc, stp, B, C,
                                         (float)ND, outPos, outMomN, Sfac);

  // 3) streaming scale of mom
  const long total = (long)B * ND;
  if ((ND & 3) == 0) {
    const long tv = total / 4;
    const int blocks = (int)((tv + TPB - 1) / TPB);
    nhc_scale_vec4<<<blocks, TPB, 0, stream>>>((const float4*)mom,
                                               (float4*)outMom, Sfac,
                                               ND / 4, tv);
  } else {
    const int blocks = (int)((total + TPB - 1) / TPB);
    nhc_scale_scalar<<<blocks, TPB, 0, stream>>>(mom, outMom, Sfac, ND, total);
  }
}